// Conv_agg_73916387164332
// MI455X (gfx1250) — compile-verified
//
#include <hip/hip_runtime.h>

typedef float v2f __attribute__((ext_vector_type(2)));
typedef float v4f __attribute__((ext_vector_type(4)));
typedef float v8f __attribute__((ext_vector_type(8)));

#define CIN   128
#define COUT  128
#define KDIM  2
#define DEG   16
#define KC    (KDIM * CIN)   // 256: reduction dimension of the fused GEMM
#define NPB   16             // nodes per block (one 16-row WMMA tile)
#define EPB   (NPB * DEG)    // 256 edges per block
#define AGG_STRIDE 260       // 256 + 4 pad: column reads hit distinct LDS banks

// Fused: (edge gather + X-weighted aggregate -> LDS) then
//        out[16,128] = agg[16,256] @ W2[256,128] + bias  via V_WMMA_F32_16X16X4_F32
__global__ __launch_bounds__(256)
void conv_agg_fused_kernel(const float* __restrict__ h,
                           const float* __restrict__ X,
                           const long long* __restrict__ edge_index,
                           const float* __restrict__ weight,   // [K][CIN][COUT]
                           const float* __restrict__ bias,     // [COUT]
                           float* __restrict__ out,            // [NT][COUT]
                           long long E_total)
{
    __shared__ float s_agg[NPB * AGG_STRIDE];  // [node][k*128 + c]
    __shared__ int   s_tgt[EPB];
    __shared__ v2f   s_x[EPB];

    const int tid  = threadIdx.x;
    const int lane = tid & 31;
    const int wave = tid >> 5;
    const long long nodeBase = (long long)blockIdx.x * NPB;
    const long long eBase    = nodeBase * DEG;

    // Weight (128 KB) is reused by every block: nudge it into cache early.
    __builtin_prefetch(weight + tid * 32, 0, 1);

    // -------- Phase 1: edge metadata -> LDS (one edge per thread) --------
    {
        const long long e = eBase + tid;
        s_tgt[tid] = (int)edge_index[2 * E_total + e];       // row 2 = global tgt id
        s_x[tid]   = *(const v2f*)(X + 2 * e);               // X[e, 0..1]
    }
    __syncthreads();

    // -------- Phase 2: aggregation; wave owns 2 nodes, lane owns 4 channels ----
    {
        const int c = lane * 4;
        for (int ln = wave * 2; ln < wave * 2 + 2; ++ln) {
            v4f acc0 = {0.f, 0.f, 0.f, 0.f};
            v4f acc1 = {0.f, 0.f, 0.f, 0.f};
            #pragma unroll
            for (int d = 0; d < DEG; ++d) {
                const int idx = ln * DEG + d;
                const int tg  = s_tgt[idx];          // wave-uniform (LDS broadcast)
                const v2f x   = s_x[idx];
                const v4f hv  = *(const v4f*)(h + (long long)tg * CIN + c);
                acc0 += x.x * hv;                    // k = 0
                acc1 += x.y * hv;                    // k = 1
            }
            *(v4f*)(&s_agg[ln * AGG_STRIDE + c])       = acc0;
            *(v4f*)(&s_agg[ln * AGG_STRIDE + CIN + c]) = acc1;
        }
    }
    __syncthreads();

    // -------- Phase 3: WMMA GEMM; wave w -> output cols [16w, 16w+16) --------
    {
        const int n0   = wave * 16;
        const int half = lane >> 4;     // 0: K rows {0,1}; 1: K rows {2,3} of each step
        const int l16  = lane & 15;

        v8f c8 = {0.f, 0.f, 0.f, 0.f, 0.f, 0.f, 0.f, 0.f};

        #pragma unroll 4
        for (int kk = 0; kk < KC / 4; ++kk) {
            const int k4 = kk * 4 + half * 2;

            // A fragment: 16x4 tile of agg. VGPR0/1 <-> K pair per 16-lane half.
            v2f a;
            a.x = s_agg[l16 * AGG_STRIDE + k4];
            a.y = s_agg[l16 * AGG_STRIDE + k4 + 1];

            // B fragment: 4x16 tile of W2, W2[kc, o] = weight[(kc>>7)*CIN*COUT + (kc&127)*COUT + o]
            const int kc0 = k4, kc1 = k4 + 1;
            v2f b;
            b.x = weight[((kc0 >> 7) * CIN + (kc0 & 127)) * COUT + n0 + l16];
            b.y = weight[((kc1 >> 7) * CIN + (kc1 & 127)) * COUT + n0 + l16];

            // D = A x B + C   (8 args: neg_a, A, neg_b, B, c_mod, C, reuse_a, reuse_b)
            c8 = __builtin_amdgcn_wmma_f32_16x16x4_f32(false, a, false, b,
                                                       (short)0, c8, false, false);
        }

        // C/D layout: VGPR j -> row M = j (+8 for lanes 16..31), col = lane&15.
        const float bv = bias[n0 + l16];
        #pragma unroll
        for (int j = 0; j < 8; ++j) {
            const int m = j + half * 8;
            out[(nodeBase + m) * COUT + n0 + l16] = c8[j] + bv;
        }
    }
}

extern "C" void kernel_launch(void* const* d_in, const int* in_sizes, int n_in,
                              void* d_out, int out_size, void* d_ws, size_t ws_size,
                              hipStream_t stream) {
    // setup_inputs order: h, X, edge_index, node_index, batch_node, batch_edge,
    //                     num_node, weight, bias
    const float*     h      = (const float*)d_in[0];
    const float*     X      = (const float*)d_in[1];
    const long long* eidx   = (const long long*)d_in[2];   // int64 [3, E]
    const float*     weight = (const float*)d_in[7];       // [K, CIN, COUT]
    const float*     bias   = (const float*)d_in[8];       // [COUT]
    float*           out    = (float*)d_out;

    const long long E  = (long long)in_sizes[2] / 3;   // 1,600,000
    const int       NT = in_sizes[0] / CIN;            // 100,000
    const int       nBlocks = NT / NPB;                // 6,250

    conv_agg_fused_kernel<<<nBlocks, 256, 0, stream>>>(h, X, eidx, weight, bias,
                                                       out, E);
}